// PDEFunc_36816459662156
// MI455X (gfx1250) — compile-verified
//
#include <hip/hip_runtime.h>

// ---------------------------------------------------------------------------
// Reference: K=32, D=96, MB=32, NT=128, NX*NY=4096.
// Key algebra: x0 is a broadcast and h0=0  =>  all 32 batch rows are
// identical for every step; and since x==h after step 0 and the r/z gates
// only consume (i_*+h_*), the per-step GEMM collapses to a 96 -> 384 matvec
// with fused weights:
//   cols [0,96):    W_ih[0:96)   + W_hh[0:96)     (r-gate, summed)
//   cols [96,192):  W_ih[96:192) + W_hh[96:192)   (z-gate, summed)
//   cols [192,288): W_ih[192:288)                 (i_n)
//   cols [288,384): W_hh[192:288)                 (h_n)
// Step 0 fits the same formula with the W_hh pre-activation forced to 0.
// ---------------------------------------------------------------------------
#define Dh    96
#define MBs   32
#define NTs   128
#define Kb    32
#define NXY   4096

typedef float v2f __attribute__((ext_vector_type(2)));
typedef float v8f __attribute__((ext_vector_type(8)));

// ===========================================================================
// Kernel 1: sequential GRU as a 128-step vector recurrence.
// One workgroup, 768 threads = 24 wave32s; wave w owns n-tile w (16 of the
// 384 fused-gate columns).  Its fused weight tile (24 ksteps x v2f = 48
// VGPRs) is built once and stays register-resident across all steps, so the
// inner loop is: 24 broadcast ds_loads of h + 24 chained v_wmma_f32_16x16x4.
// A-tile rows are all equal to h (broadcast), so C row 0 is the matvec.
// ===========================================================================
__global__ __launch_bounds__(768) void gru_seq_kernel(
    const float* __restrict__ init_coeffs,   // (96)
    const float* __restrict__ W_ih,          // (288, 96)
    const float* __restrict__ W_hh,          // (288, 96)
    const float* __restrict__ b_ih,          // (288)
    const float* __restrict__ b_hh,          // (288)
    float* __restrict__ coeff_ws)            // (3, 128, 32)
{
  __shared__ __attribute__((aligned(16))) float G[384];     // fused pre-acts
  __shared__ __attribute__((aligned(16))) float hb[2][Dh];  // h ping-pong

  const int tid  = threadIdx.x;
  const int lane = tid & 31;
  const int mloc = lane & 15;
  const int koff = (lane >> 4) * 2;    // K sub-offset of the WMMA layout
  const int ntile = tid >> 5;          // 0..23: this wave's 16-column tile

  // ---- build this wave's fused weight tile in registers (B operand) ----
  const int grp  = ntile / 6;                 // 0:r 1:z 2:i_n 3:h_n
  const int ncol = (ntile % 6) * 16 + mloc;   // 0..95 within the group
  const int wrow = (grp == 0 ? 0 : (grp == 1 ? 96 : 192)) + ncol;
  const float* p_ih = W_ih + wrow * Dh;
  const float* p_hh = W_hh + wrow * Dh;
  v2f b[24];
#pragma unroll
  for (int ks = 0; ks < 24; ++ks) {
    const int k = 4 * ks + koff;
    v2f v;
    if (grp <= 1) {            // summed r/z weights
      v2f u = *(const v2f*)(p_ih + k);
      v2f w = *(const v2f*)(p_hh + k);
      v.x = u.x + w.x; v.y = u.y + w.y;
    } else if (grp == 2) {     // i_n
      v = *(const v2f*)(p_ih + k);
    } else {                   // h_n
      v = *(const v2f*)(p_hh + k);
    }
    b[ks] = v;
  }

  // ---- fused biases (threads 0..95 own gate element d) ----
  float br = 0.f, bz = 0.f, bin = 0.f, bhn = 0.f;
  if (tid < Dh) {
    br  = b_ih[tid]       + b_hh[tid];
    bz  = b_ih[96 + tid]  + b_hh[96 + tid];
    bin = b_ih[192 + tid];
    bhn = b_hh[192 + tid];
    hb[0][tid] = 0.0f;                        // h0 = 0
  }

  // ---- step-0 pre-activations: G[0:288) = W_ih @ x0, G[288:384) = 0 ----
  if (tid < 288) {
    const float* wr = W_ih + tid * Dh;
    float s0 = 0.f, s1 = 0.f, s2 = 0.f, s3 = 0.f;
#pragma unroll
    for (int k = 0; k < Dh; k += 4) {
      s0 += init_coeffs[k]     * wr[k];
      s1 += init_coeffs[k + 1] * wr[k + 1];
      s2 += init_coeffs[k + 2] * wr[k + 2];
      s3 += init_coeffs[k + 3] * wr[k + 3];
    }
    G[tid] = (s0 + s1) + (s2 + s3);
  } else if (tid < 384) {
    G[tid] = 0.0f;                            // W_hh @ h0 == 0
  }

  for (int t = 0; t < NTs; ++t) {
    if (t > 0) {
      // ---- matvec via WMMA: A rows all = h (broadcast loads from LDS) ----
      const float* hcur = hb[t & 1];
      v8f acc0 = {}, acc1 = {};
#pragma unroll
      for (int ks = 0; ks < 24; ks += 2) {    // two chains hide C-RAW latency
        v2f a0 = *(const v2f*)(hcur + 4 * ks + koff);
        v2f a1 = *(const v2f*)(hcur + 4 * (ks + 1) + koff);
        acc0 = __builtin_amdgcn_wmma_f32_16x16x4_f32(
            false, a0, false, b[ks],     (short)0, acc0, false, false);
        acc1 = __builtin_amdgcn_wmma_f32_16x16x4_f32(
            false, a1, false, b[ks + 1], (short)0, acc1, false, false);
      }
      if (lane < 16)                          // C row 0 lives in vgpr0,lanes 0-15
        G[ntile * 16 + mloc] = acc0[0] + acc1[0];
    }
    __syncthreads();

    // ---- gate nonlinearity on threads 0..95 ----
    if (tid < Dh) {
      const int d = tid;
      float r = 1.0f / (1.0f + __expf(-(G[d] + br)));
      float z = 1.0f / (1.0f + __expf(-(G[96 + d] + bz)));
      float n = tanhf(G[192 + d] + bin + r * (G[288 + d] + bhn));
      float hnew = (1.0f - z) * n + z * hb[t & 1][d];
      hb[(t + 1) & 1][d] = hnew;
      const int k = d / 3, c = d % 3;         // coeff.reshape(.., K, 3)
      coeff_ws[c * (NTs * Kb) + t * Kb + k] = hnew;
    }
    __syncthreads();
  }
}

// ===========================================================================
// Kernel 2: plane[t, c, n] = coeff_c[t, 0:32] @ basis_c[0:32, n], then the
// value is replicated to all 32 (identical) batch slots at store time.
// Per channel: GEMM M=128 (t), N=4096, K=32 via fp32 WMMA (8 k-steps).
// Pure streaming store of 201 MB -> HBM-write-bound (~8.6 us at 23.3 TB/s);
// basis (1.5 MB) and coeff (48 KB) stay L2-resident.
// Block = 128 threads (4 waves); wave -> one 16x16 (t x n) tile.
// ===========================================================================
__global__ __launch_bounds__(128) void basis_expand_kernel(
    const float* __restrict__ coeff_ws,   // (3, 128, 32)
    const float* __restrict__ basis,      // (32, 3, 4096)
    float* __restrict__ out)              // (128, 32, 3, 4096)
{
  const int c     = blockIdx.z;
  const int tbase = blockIdx.y * 16;
  const int tid   = threadIdx.x;
  const int wave  = tid >> 5;
  const int lane  = tid & 31;
  const int mloc  = lane & 15;
  const int koff  = (lane >> 4) * 2;
  const int n     = (blockIdx.x * 4 + wave) * 16 + mloc;

  const float* __restrict__ A = coeff_ws + c * (NTs * Kb);

  v2f a[8];
#pragma unroll
  for (int ks = 0; ks < 8; ++ks)
    a[ks] = *(const v2f*)(A + (tbase + mloc) * Kb + 4 * ks + koff);

  v8f acc0 = {}, acc1 = {};
#pragma unroll
  for (int ks = 0; ks < 8; ks += 2) {
    const int k0 = 4 * ks + koff;
    const int k1 = 4 * (ks + 1) + koff;
    v2f b0, b1;
    b0.x = basis[(k0 * 3 + c) * NXY + n];          // B[k][n], k-stride 3*4096
    b0.y = basis[((k0 + 1) * 3 + c) * NXY + n];
    b1.x = basis[(k1 * 3 + c) * NXY + n];
    b1.y = basis[((k1 + 1) * 3 + c) * NXY + n];
    acc0 = __builtin_amdgcn_wmma_f32_16x16x4_f32(
        false, a[ks],     false, b0, (short)0, acc0, false, false);
    acc1 = __builtin_amdgcn_wmma_f32_16x16x4_f32(
        false, a[ks + 1], false, b1, (short)0, acc1, false, false);
  }

#pragma unroll
  for (int r = 0; r < 8; ++r) {
    const int t = tbase + r + (lane >> 4) * 8;
    const float val = acc0[r] + acc1[r];
    float* po = out + ((size_t)(t * MBs) * 3 + c) * NXY + n;
#pragma unroll
    for (int m = 0; m < MBs; ++m)                  // identical batch rows
      po[(size_t)m * 3 * NXY] = val;               // imm-offset b32 stores
  }
}

// ===========================================================================
extern "C" void kernel_launch(void* const* d_in, const int* in_sizes, int n_in,
                              void* d_out, int out_size, void* d_ws, size_t ws_size,
                              hipStream_t stream) {
  // inputs: 0=grid0 (unused), 1=t (length only), 2=init_coeffs, 3=basis,
  //         4=W_ih, 5=W_hh, 6=b_ih, 7=b_hh
  const float* init_coeffs = (const float*)d_in[2];
  const float* basis       = (const float*)d_in[3];
  const float* W_ih        = (const float*)d_in[4];
  const float* W_hh        = (const float*)d_in[5];
  const float* b_ih        = (const float*)d_in[6];
  const float* b_hh        = (const float*)d_in[7];
  float* out = (float*)d_out;
  float* ws  = (float*)d_ws;   // 3*128*32 floats = 48 KB

  gru_seq_kernel<<<dim3(1), dim3(768), 0, stream>>>(
      init_coeffs, W_ih, W_hh, b_ih, b_hh, ws);

  basis_expand_kernel<<<dim3(NXY / 64, NTs / 16, 3), dim3(128), 0, stream>>>(
      ws, basis, out);
}